// RetentionChannel_77867757077203
// MI455X (gfx1250) — compile-verified
//
#include <hip/hip_runtime.h>

// Retention (RetNet-style chunked scan), f32, CDNA5 WMMA 16x16x4 f32 path.
// B=4 S=8192 H=16 D=E=64 CHUNK=128. One workgroup (512 thr = 16 wave32) per
// (b,h); serial scan over 64 chunks; state (64x64 f32) kept in LDS.
// v6: per-position decay/mask products (pref*mf, inv*mf, pref_last*inv*mf)
// precomputed once per chunk into LDS, so fragment scaling is one packed
// multiply instead of chained v_pk_mul per fragment. Otherwise v5: scalar
// wave-uniform control, template<NT> Phase A, 2x interleaved inner loops,
// prefetch issued before Phase C to keep Phases A/B at low VGPR pressure.

#define BB 4
#define SS 8192
#define HH 16
#define DD 64
#define EE 64
#define CC 128
#define NCH (SS / CC)
#define EPSF 1e-8f

typedef float v2f __attribute__((ext_vector_type(2)));
typedef float v4f __attribute__((ext_vector_type(4)));
typedef float v8f __attribute__((ext_vector_type(8)));

// LDS strides (floats).
//  LQ mod 64 == 12: conflict-free for (a) per-lane-row reads with K-pair
//  split across halves and (b) column reads with rows k0/k0+2 across halves.
//  LW mod 64 == 4:  conflict-free for row-major A reads of the weights tile.
//  Both strides*4B are multiples of 16B so float4 staging stays aligned.
#define LQ 76
#define LW 132

// LDS layout (float offsets)
#define OFF_Q 0
#define OFF_K (OFF_Q + CC * LQ)
#define OFF_V (OFF_K + CC * LQ)
#define OFF_W (OFF_V + CC * LQ)
#define OFF_S (OFF_W + CC * LW)     // state 64 x LQ
#define OFF_PREF (OFF_S + DD * LQ)  // 128: cumprod(decay) (scan in place)
#define OFF_MF (OFF_PREF + CC)      // 128: mask factor
#define OFF_PM (OFF_MF + CC)        // 128: pref[c]*mf[c]
#define OFF_IM (OFF_PM + CC)        // 128: inv[m]*mf[m]
#define OFF_CW (OFF_IM + CC)        // 128: pref_last*inv[m]*mf[m]
#define OFF_G (OFF_CW + CC)         // 1 (+pad)
#define SMEM_FLOATS (OFF_G + 4)     // ~206 KB < 320 KB/WG

#define ZACC {0.f, 0.f, 0.f, 0.f, 0.f, 0.f, 0.f, 0.f}

__device__ __forceinline__ v8f wmma4(v2f a, v2f b, v8f c) {
  // V_WMMA_F32_16X16X4_F32: D = A(16x4) * B(4x16) + C(16x16), wave32.
  return __builtin_amdgcn_wmma_f32_16x16x4_f32(false, a, false, b, (short)0, c,
                                               false, false);
}

// Phase A body with a compile-time number of active (causal) K tiles.
// All WMMAs unconditional -> no exec masking, clean ds/wmma pipelining.
template <int NT>
__device__ __forceinline__ void phaseA_impl(
    const float* __restrict__ sQ, const float* __restrict__ sK,
    float* __restrict__ sW, const float* __restrict__ sPM,
    const float* __restrict__ sIM, int r, int mb0, int half, int lr) {
  const v8f z = ZACC;
  v8f acc[NT];
#pragma unroll
  for (int i = 0; i < NT; ++i) acc[i] = z;
  const float* Arow = sQ + (r * 16 + lr) * LQ;
  const float* Brow[NT];
#pragma unroll
  for (int i = 0; i < NT; ++i) Brow[i] = sK + ((mb0 + i) * 16 + lr) * LQ;
#pragma unroll
  for (int kk = 0; kk < 16; ++kk) {
    const int k0 = kk * 4 + 2 * half;
    v2f a;
    a.x = Arow[k0];
    a.y = Arow[k0 + 1];
    v2f bm[NT];
#pragma unroll
    for (int i = 0; i < NT; ++i) {
      bm[i].x = Brow[i][k0];
      bm[i].y = Brow[i][k0 + 1];
    }
#pragma unroll
    for (int i = 0; i < NT; ++i) acc[i] = wmma4(a, bm[i], acc[i]);
  }
#pragma unroll
  for (int i = 0; i < NT; ++i) {
    const int m = (mb0 + i) * 16 + lr;
    const float fm = sIM[m];  // inv[m]*mf[m] (k-mask covers v-mask: mf^2=mf)
#pragma unroll
    for (int j = 0; j < 8; ++j) {
      const int c = r * 16 + half * 8 + j;
      const float f = (m <= c) ? sPM[c] * fm : 0.f;
      sW[c * LW + m] = acc[i][j] * f;
    }
  }
}

__global__ __launch_bounds__(512) void retention_wmma_kernel(
    const float* __restrict__ q, const float* __restrict__ k,
    const float* __restrict__ v, const unsigned char* __restrict__ mask,
    const float* __restrict__ gp, float* __restrict__ out) {
  extern __shared__ float sm[];
  float* sQ = sm + OFF_Q;
  float* sK = sm + OFF_K;
  float* sV = sm + OFF_V;
  float* sW = sm + OFF_W;
  float* sS = sm + OFF_S;
  float* sPREF = sm + OFF_PREF;
  float* sMF = sm + OFF_MF;
  float* sPM = sm + OFF_PM;
  float* sIM = sm + OFF_IM;
  float* sCW = sm + OFF_CW;

  const int t = threadIdx.x;
  // wave id is uniform per wave: force it into an SGPR so everything derived
  // from it (tile assignment, loop bounds) gets scalar branches.
  const int wave = __builtin_amdgcn_readfirstlane(t >> 5);
  const int lane = t & 31;
  const int half = lane >> 4;  // K-pair select for A/B frags, row-half for C/D
  const int lr = lane & 15;
  const int bh = blockIdx.x;
  const int bb = bh / HH;
  const int hh = bh % HH;

  // gamma_h = exp(-cumsum(softplus(gamma_param))[h])
  if (t == 0) {
    float acc = 0.f;
    for (int j = 0; j <= hh; ++j) {
      float x = gp[j];
      acc += (x > 20.f) ? x : log1pf(__expf(x));
    }
    sm[OFF_G] = __expf(-acc);
  }
  // zero the running state
  for (int i = t; i < DD * LQ; i += 512) sS[i] = 0.f;
  __syncthreads();
  const float gamma = sm[OFF_G];

  // ---- register-prefetch pipeline state ----
  v4f rq[4], rk[4], rv[4];
  unsigned char rmb = 0;

  auto prefetch = [&](int s0) {
#pragma unroll
    for (int i = 0; i < 4; ++i) {
      const int idx = t + i * 512;  // 0..2047
      const int row = idx >> 4;     // 0..127
      const int c4 = (idx & 15) << 2;
      const size_t gofs = (((size_t)(bb * SS + s0 + row)) * HH + hh) * DD + c4;
      rq[i] = __builtin_nontemporal_load((const v4f*)(q + gofs));
      rk[i] = __builtin_nontemporal_load((const v4f*)(k + gofs));
      rv[i] = __builtin_nontemporal_load((const v4f*)(v + gofs));
    }
    if (t < CC) rmb = mask[(size_t)bb * SS + s0 + t];
  };
  prefetch(0);

  for (int nc = 0; nc < NCH; ++nc) {
    const int s0 = nc * CC;

    // ---- stage prefetched chunk into LDS (raw; mask folded in later) ----
#pragma unroll
    for (int i = 0; i < 4; ++i) {
      const int idx = t + i * 512;
      const int row = idx >> 4;
      const int c4 = (idx & 15) << 2;
      *(v4f*)(sQ + row * LQ + c4) = rq[i];
      *(v4f*)(sK + row * LQ + c4) = rk[i];
      *(v4f*)(sV + row * LQ + c4) = rv[i];
    }
    if (t < CC) {
      sPREF[t] = fmaxf(rmb ? gamma : 1.f, EPSF);  // scan input (in place)
      sMF[t] = rmb ? 1.f : 0.f;
    }
    __syncthreads();

    // ---- inclusive cumprod of decay (Hillis-Steele), ref f32 semantics ----
    for (int off = 1; off < CC; off <<= 1) {
      float val = 1.f;
      const bool act = (t < CC) && (t >= off);
      if (act) val = sPREF[t - off];
      __syncthreads();
      if (act) sPREF[t] *= val;
      __syncthreads();
    }
    // per-position products used by all phases, computed once per chunk
    if (t < CC) {
      const float p = sPREF[t];
      const float inv = 1.f / fmaxf(p, EPSF);
      const float mf = sMF[t];
      sPM[t] = p * mf;                      // pref[c]*mf[c]
      sIM[t] = inv * mf;                    // inv[m]*mf[m]
      sCW[t] = sPREF[CC - 1] * inv * mf;    // pref_last*inv[m]*mf[m]
    }
    __syncthreads();

    // ---- Phase A: W = (Q @ K^T) * decay * causal * mask ----
    // One A (Q) fragment feeds up to 4 K tiles; NT dispatched at compile time.
    {
      const int r = wave >> 1;         // row block 0..7 (SGPR)
      const int mb0 = (wave & 1) * 4;  // col-block half (SGPR)
      int nt = r - mb0 + 1;            // active (causal) tiles in this half
      if (nt > 4) nt = 4;
      if (nt == 4) {
        phaseA_impl<4>(sQ, sK, sW, sPM, sIM, r, mb0, half, lr);
      } else if (nt == 3) {
        phaseA_impl<3>(sQ, sK, sW, sPM, sIM, r, mb0, half, lr);
      } else if (nt == 2) {
        phaseA_impl<2>(sQ, sK, sW, sPM, sIM, r, mb0, half, lr);
      } else if (nt == 1) {
        phaseA_impl<1>(sQ, sK, sW, sPM, sIM, r, mb0, half, lr);
      }
    }
    __syncthreads();

    // ---- Phase B: out = ((Q*prefix*mask) @ state + W @ V) * mask ----
    // One A fragment feeds both E tiles; 2x-interleaved fragment loads.
    {
      const int r = wave >> 1;         // SGPR
      const int eb0 = (wave & 1) * 2;  // SGPR
      v8f acc0 = ZACC, acc1 = ZACC;
      // prev term: K over d (16 steps, 2x interleaved)
      const float ps = sPM[r * 16 + lr];
      const float* Arow = sQ + (r * 16 + lr) * LQ;
#pragma unroll
      for (int kk = 0; kk < 16; kk += 2) {
        const int ka = kk * 4 + 2 * half;
        const int kb = ka + 4;
        v2f aA, aB;
        aA.x = Arow[ka] * ps;
        aA.y = Arow[ka + 1] * ps;
        aB.x = Arow[kb] * ps;
        aB.y = Arow[kb + 1] * ps;
        v2f b00, b01, b10, b11;
        b00.x = sS[(ka + 0) * LQ + (eb0 + 0) * 16 + lr];
        b00.y = sS[(ka + 1) * LQ + (eb0 + 0) * 16 + lr];
        b01.x = sS[(ka + 0) * LQ + (eb0 + 1) * 16 + lr];
        b01.y = sS[(ka + 1) * LQ + (eb0 + 1) * 16 + lr];
        b10.x = sS[(kb + 0) * LQ + (eb0 + 0) * 16 + lr];
        b10.y = sS[(kb + 1) * LQ + (eb0 + 0) * 16 + lr];
        b11.x = sS[(kb + 0) * LQ + (eb0 + 1) * 16 + lr];
        b11.y = sS[(kb + 1) * LQ + (eb0 + 1) * 16 + lr];
        acc0 = wmma4(aA, b00, acc0);
        acc1 = wmma4(aA, b01, acc1);
        acc0 = wmma4(aB, b10, acc0);
        acc1 = wmma4(aB, b11, acc1);
      }
      // intra term: K over m, truncated at the causal bound (scalar bound,
      // always even -> 2x interleaved)
      const float* Wrow = sW + (r * 16 + lr) * LW;
      const int kmax = 4 * (r + 1);
      for (int kk = 0; kk < kmax; kk += 2) {
        const int ka = kk * 4 + 2 * half;
        const int kb = ka + 4;
        v2f aA, aB;
        aA.x = Wrow[ka];
        aA.y = Wrow[ka + 1];
        aB.x = Wrow[kb];
        aB.y = Wrow[kb + 1];
        v2f b00, b01, b10, b11;
        b00.x = sV[(ka + 0) * LQ + (eb0 + 0) * 16 + lr];
        b00.y = sV[(ka + 1) * LQ + (eb0 + 0) * 16 + lr];
        b01.x = sV[(ka + 0) * LQ + (eb0 + 1) * 16 + lr];
        b01.y = sV[(ka + 1) * LQ + (eb0 + 1) * 16 + lr];
        b10.x = sV[(kb + 0) * LQ + (eb0 + 0) * 16 + lr];
        b10.y = sV[(kb + 1) * LQ + (eb0 + 0) * 16 + lr];
        b11.x = sV[(kb + 0) * LQ + (eb0 + 1) * 16 + lr];
        b11.y = sV[(kb + 1) * LQ + (eb0 + 1) * 16 + lr];
        acc0 = wmma4(aA, b00, acc0);
        acc1 = wmma4(aA, b01, acc1);
        acc0 = wmma4(aB, b10, acc0);
        acc1 = wmma4(aB, b11, acc1);
      }
#pragma unroll
      for (int i = 0; i < 8; ++i) {
        const int c = r * 16 + half * 8 + i;
        const float mm = sMF[c];
        const size_t o0 =
            (((size_t)(bb * SS + s0 + c)) * HH + hh) * EE + eb0 * 16 + lr;
        __builtin_nontemporal_store(acc0[i] * mm, out + o0);
        __builtin_nontemporal_store(acc1[i] * mm, out + o0 + 16);
      }
    }
    __syncthreads();  // all reads of old state complete before updates

    // kick off next chunk's HBM traffic here: covered by Phase C + loop tail,
    // and keeps the 48 prefetch VGPRs dead during Phases A/B.
    if (nc + 1 < NCH) prefetch(s0 + CC);

    // ---- Phase C: state = state*prefix_last + (K ⊙ cw)^T @ V ----
    // cw[m] = pref_last*inv[m]*mf[m] precomputed; one packed mul per frag.
    {
      const int db = wave >> 2;  // 0..3 (SGPR)
      const int eb = wave & 3;   // 0..3 (SGPR)
      const float pl = sPREF[CC - 1];
      v8f acc = ZACC;
#pragma unroll
      for (int kk = 0; kk < 32; kk += 2) {
        const int ka = kk * 4 + 2 * half;
        const int kb = ka + 4;
        v2f cwA, cwB;
        cwA.x = sCW[ka];
        cwA.y = sCW[ka + 1];
        cwB.x = sCW[kb];
        cwB.y = sCW[kb + 1];
        v2f kfA, kfB;
        kfA.x = sK[(ka + 0) * LQ + db * 16 + lr];
        kfA.y = sK[(ka + 1) * LQ + db * 16 + lr];
        kfB.x = sK[(kb + 0) * LQ + db * 16 + lr];
        kfB.y = sK[(kb + 1) * LQ + db * 16 + lr];
        v2f bA, bB;
        bA.x = sV[(ka + 0) * LQ + eb * 16 + lr];
        bA.y = sV[(ka + 1) * LQ + eb * 16 + lr];
        bB.x = sV[(kb + 0) * LQ + eb * 16 + lr];
        bB.y = sV[(kb + 1) * LQ + eb * 16 + lr];
        acc = wmma4(kfA * cwA, bA, acc);
        acc = wmma4(kfB * cwB, bB, acc);
      }
#pragma unroll
      for (int i = 0; i < 8; ++i) {
        const int d = db * 16 + half * 8 + i;
        const int e = eb * 16 + lr;
        const int idx = d * LQ + e;
        sS[idx] = sS[idx] * pl + acc[i];  // each wave owns a disjoint tile
      }
    }
    __syncthreads();  // state + LDS tiles stable before next chunk
  }
}

extern "C" void kernel_launch(void* const* d_in, const int* in_sizes, int n_in,
                              void* d_out, int out_size, void* d_ws,
                              size_t ws_size, hipStream_t stream) {
  const float* q = (const float*)d_in[0];
  const float* k = (const float*)d_in[1];
  const float* v = (const float*)d_in[2];
  const unsigned char* mask = (const unsigned char*)d_in[3];  // jnp bool = u8
  const float* gp = (const float*)d_in[4];
  float* out = (float*)d_out;

  const size_t smem = (size_t)SMEM_FLOATS * sizeof(float);  // ~206 KB < 320 KB
  (void)hipFuncSetAttribute((const void*)retention_wmma_kernel,
                            hipFuncAttributeMaxDynamicSharedMemorySize,
                            (int)smem);

  retention_wmma_kernel<<<dim3(BB * HH), dim3(512), smem, stream>>>(
      q, k, v, mask, gp, out);
}